// ForcePai_68453188763754
// MI455X (gfx1250) — compile-verified
//
#include <hip/hip_runtime.h>
#include <math.h>

#define FEAT   128
#define NRBF   20
#define NCONV  3
#define PI_F   3.14159265358979323846f
#define CUT_F  5.0f

typedef float v2f __attribute__((ext_vector_type(2)));
typedef float v8f __attribute__((ext_vector_type(8)));

// ---------------------------------------------------------------- utilities
__global__ void zero_kernel(float* __restrict__ p, int n) {
    int i = blockIdx.x * blockDim.x + threadIdx.x;
    if (i < n) p[i] = 0.0f;
}

__global__ void embed_kernel(float* __restrict__ s, const float* __restrict__ embed,
                             const int* __restrict__ z, int N) {
    int i = blockIdx.x * blockDim.x + threadIdx.x;
    if (i < N * FEAT) s[i] = embed[z[i >> 7] * FEAT + (i & 127)];
}

// --------------------------------------------------------------- geometry
__global__ void geom_kernel(const float* __restrict__ xyz, const int* __restrict__ nbr,
                            float* __restrict__ dist, float* __restrict__ unit,
                            float* __restrict__ env, int E) {
    int e = blockIdx.x * blockDim.x + threadIdx.x;
    if (e >= E) return;
    int i = nbr[2 * e + 0];
    int j = nbr[2 * e + 1];
    float rx = xyz[3 * j + 0] - xyz[3 * i + 0];
    float ry = xyz[3 * j + 1] - xyz[3 * i + 1];
    float rz = xyz[3 * j + 2] - xyz[3 * i + 2];
    // reference adds EPS per component before summing: sum + 3*EPS
    float d = sqrtf(rx * rx + ry * ry + rz * rz + 3.0e-15f);
    dist[e] = d;
    unit[3 * e + 0] = rx / d;
    unit[3 * e + 1] = ry / d;
    unit[3 * e + 2] = rz / d;
    env[e] = (d < CUT_F) ? 0.5f * (cosf(PI_F * d / CUT_F) + 1.0f) : 0.0f;
}

// ------------------------------------------------------- WMMA f32 GEMM
// C[M,N] = act(A[M,K] @ W[K,N] + bias).
// One wave computes MT vertically-stacked 16x16 tiles, reusing the B fragment
// across all MT WMMAs per K-step (B-load amortization).
// grid = (M/(16*MT), N/16, batch); block = 32.  All dims multiples of 16/4;
// no divergence before WMMA -> EXEC all-ones as required.
template <int MT>
__global__ void gemm_wmma(const float* __restrict__ A, const float* __restrict__ W,
                          const float* __restrict__ bias, float* __restrict__ C,
                          int K, int N, int act_silu, int aStride, int cStride) {
    int lane = threadIdx.x & 31;
    int half = lane >> 4;        // 0: K 0..1 | 1: K 2..3   (A / B fragment halves)
    int l16  = lane & 15;
    const float* Ab = A + (size_t)blockIdx.z * aStride;
    float*       Cb = C + (size_t)blockIdx.z * cStride;
    int m0 = blockIdx.x * (16 * MT);
    int n0 = blockIdx.y * 16;

    const float* arow[MT];
#pragma unroll
    for (int t = 0; t < MT; ++t)
        arow[t] = Ab + (size_t)(m0 + t * 16 + l16) * K + half * 2;  // 8B-aligned (K%4==0)
    const float* bcol = W + (n0 + l16);

    v8f acc[MT];
#pragma unroll
    for (int t = 0; t < MT; ++t) acc[t] = (v8f){};

    for (int k0 = 0; k0 < K; k0 += 4) {
        v2f b;
        b.x = bcol[(size_t)(k0 + half * 2 + 0) * N];
        b.y = bcol[(size_t)(k0 + half * 2 + 1) * N];
#pragma unroll
        for (int t = 0; t < MT; ++t) {
            v2f a = *(const v2f*)(arow[t] + k0);   // global_load_b64
            acc[t] = __builtin_amdgcn_wmma_f32_16x16x4_f32(
                         /*neg_a=*/false, a, /*neg_b=*/false, b,
                         /*c_mod=*/(short)0, acc[t],
                         /*reuse_a=*/false, /*reuse_b=*/false);
        }
    }
    // D layout: VGPR r, lanes 0-15 -> (M=r, N=lane); lanes 16-31 -> (M=8+r, N=lane-16)
#pragma unroll
    for (int t = 0; t < MT; ++t) {
#pragma unroll
        for (int r = 0; r < 8; ++r) {
            int m = m0 + t * 16 + r + half * 8;
            int n = n0 + l16;
            float x = acc[t][r];
            if (bias) x += bias[n];
            if (act_silu) x = x / (1.0f + expf(-x));   // x * sigmoid(x)
            Cb[(size_t)m * N + n] = x;
        }
    }
}

// ------------------------------------------------ per-edge message kernel
// One wave32 per edge.  Folds w_s = (rbf @ dist_w + dist_b) * env on the fly.
__global__ void edge_msg_kernel(const int* __restrict__ nbr, const float* __restrict__ dist,
                                const float* __restrict__ unit, const float* __restrict__ env,
                                const float* __restrict__ phi, const float* __restrict__ v,
                                const float* __restrict__ dw,  const float* __restrict__ db,
                                float* __restrict__ ds, float* __restrict__ dv,
                                int E, int N) {
    int e    = (blockIdx.x * blockDim.x + threadIdx.x) >> 5;
    int lane = threadIdx.x & 31;
    if (e >= E) return;                       // uniform per wave
    int i = nbr[2 * e + 0];
    int j = nbr[2 * e + 1];

    // warm L2/L0 for the gathered phi row (1536B; 32 lanes x 48B stride covers it)
    const float* pj = phi + (size_t)j * (3 * FEAT);
    __builtin_prefetch(pj + lane * 12, 0, 3);

    float d  = dist[e];
    float ev = env[e];
    float ux = unit[3 * e + 0], uy = unit[3 * e + 1], uz = unit[3 * e + 2];

    float rbf[NRBF];
#pragma unroll
    for (int k = 0; k < NRBF; ++k)
        rbf[k] = sinf((float)(k + 1) * (PI_F / CUT_F) * d) / d;

    const int nf = N * FEAT;
#pragma unroll
    for (int t = 0; t < 4; ++t) {
        int f = lane + 32 * t;
        float w0 = db[f], w1 = db[f + FEAT], w2 = db[f + 2 * FEAT];
#pragma unroll
        for (int k = 0; k < NRBF; ++k) {
            float r = rbf[k];
            const float* row = dw + k * (3 * FEAT) + f;
            w0 += r * row[0];
            w1 += r * row[FEAT];
            w2 += r * row[2 * FEAT];
        }
        w0 *= ev; w1 *= ev; w2 *= ev;
        float inv0 = pj[f] * w0;
        float inv1 = pj[f + FEAT] * w1;
        float inv2 = pj[f + 2 * FEAT] * w2;

        size_t jf  = (size_t)j * FEAT + f;
        size_t iff = (size_t)i * FEAT + f;
        atomicAdd(&ds[iff], inv1);
        atomicAdd(&dv[0 * nf + iff], inv0 * v[0 * nf + jf] + inv2 * ux);
        atomicAdd(&dv[1 * nf + iff], inv0 * v[1 * nf + jf] + inv2 * uy);
        atomicAdd(&dv[2 * nf + iff], inv0 * v[2 * nf + jf] + inv2 * uz);
    }
}

// s += ds; v += dv; re-zero deltas for next layer
__global__ void apply_kernel(float* __restrict__ s, float* __restrict__ v,
                             float* __restrict__ ds, float* __restrict__ dv, int nf) {
    int i = blockIdx.x * blockDim.x + threadIdx.x;
    if (i < nf)     { s[i] += ds[i]; ds[i] = 0.0f; }
    if (i < 3 * nf) { v[i] += dv[i]; dv[i] = 0.0f; }
}

// s_stack = [s, |v_v|] per atom row of 256
__global__ void stack_kernel(const float* __restrict__ s, const float* __restrict__ vv,
                             float* __restrict__ sstack, int N) {
    int i = blockIdx.x * blockDim.x + threadIdx.x;
    if (i >= N * FEAT) return;
    int n = i >> 7, f = i & 127;
    int nf = N * FEAT;
    float a = vv[i], b = vv[nf + i], c = vv[2 * nf + i];
    sstack[n * 256 + f]        = s[i];
    sstack[n * 256 + FEAT + f] = sqrtf(a * a + b * b + c * c + 1.0e-15f);
}

// v += u_v * sp0 ; s += (sum_c u_v*v_v) * sp1 + sp2
__global__ void updfinal_kernel(float* __restrict__ s, float* __restrict__ v,
                                const float* __restrict__ uv, const float* __restrict__ vv,
                                const float* __restrict__ sp, int N) {
    int i = blockIdx.x * blockDim.x + threadIdx.x;
    if (i >= N * FEAT) return;
    int n = i >> 7, f = i & 127;
    int nf = N * FEAT;
    float sp0 = sp[n * 384 + f];
    float sp1 = sp[n * 384 + FEAT + f];
    float sp2 = sp[n * 384 + 2 * FEAT + f];
    float dot = 0.0f;
#pragma unroll
    for (int c = 0; c < 3; ++c) {
        float u = uv[c * nf + i];
        dot += u * vv[c * nf + i];
        v[c * nf + i] += u * sp0;
    }
    s[i] += dot * sp1 + sp2;
}

// --------------------------------------------------- readout + force scatter
// One wave32 per edge; e-features staged in LDS; 64-wide hidden MLP.
__global__ void edge_force_kernel(const int* __restrict__ nbr, const float* __restrict__ xyz,
                                  const float* __restrict__ s,
                                  const float* __restrict__ w1, const float* __restrict__ b1,
                                  const float* __restrict__ w2, const float* __restrict__ b2,
                                  float* __restrict__ out, int E) {
    __shared__ float esh[4][FEAT];            // blockDim = 128 -> 4 waves
    int wl   = threadIdx.x >> 5;
    int lane = threadIdx.x & 31;
    int e    = (blockIdx.x * blockDim.x + threadIdx.x) >> 5;
    bool valid = (e < E);
    int ec = valid ? e : 0;
    int i = nbr[2 * ec + 0];
    int j = nbr[2 * ec + 1];
#pragma unroll
    for (int t = 0; t < 4; ++t) {
        int f = lane + 32 * t;
        esh[wl][f] = s[(size_t)i * FEAT + f] + s[(size_t)j * FEAT + f];
    }
    __syncthreads();
    float partial = 0.0f;
#pragma unroll
    for (int mm = 0; mm < 2; ++mm) {
        int m = lane + 32 * mm;
        float acc = b1[m];
        for (int f = 0; f < FEAT; ++f) acc += esh[wl][f] * w1[f * 64 + m];
        float h = acc / (1.0f + expf(-acc));   // silu
        partial += h * w2[m];
    }
#pragma unroll
    for (int off = 16; off > 0; off >>= 1) partial += __shfl_xor(partial, off, 32);
    float fs = partial + b2[0];
    float dx = xyz[3 * i + 0] - xyz[3 * j + 0];
    float dy = xyz[3 * i + 1] - xyz[3 * j + 1];
    float dz = xyz[3 * i + 2] - xyz[3 * j + 2];
    float dd = sqrtf(dx * dx + dy * dy + dz * dz);
    float sc = fs / dd;
    if (valid && lane < 3) {
        float comp = (lane == 0 ? dx : (lane == 1 ? dy : dz)) * sc;
        atomicAdd(&out[3 * i + lane],  comp);
        atomicAdd(&out[3 * j + lane], -comp);
    }
}

// ================================================================ launcher
extern "C" void kernel_launch(void* const* d_in, const int* in_sizes, int n_in,
                              void* d_out, int out_size, void* d_ws, size_t ws_size,
                              hipStream_t stream) {
    const float* xyz    = (const float*)d_in[0];
    const int*   z      = (const int*)  d_in[1];
    const int*   nbr    = (const int*)  d_in[2];
    const float* embedw = (const float*)d_in[3];
    const float* msg_w1 = (const float*)d_in[4];
    const float* msg_b1 = (const float*)d_in[5];
    const float* msg_w2 = (const float*)d_in[6];
    const float* msg_b2 = (const float*)d_in[7];
    const float* dist_w = (const float*)d_in[8];
    const float* dist_b = (const float*)d_in[9];
    const float* upd_u  = (const float*)d_in[10];
    const float* upd_v  = (const float*)d_in[11];
    const float* upd_w1 = (const float*)d_in[12];
    const float* upd_b1 = (const float*)d_in[13];
    const float* upd_w2 = (const float*)d_in[14];
    const float* upd_b2 = (const float*)d_in[15];
    const float* ero_w1 = (const float*)d_in[16];
    const float* ero_b1 = (const float*)d_in[17];
    const float* ero_w2 = (const float*)d_in[18];
    const float* ero_b2 = (const float*)d_in[19];

    const int N  = in_sizes[1];          // 10000 atoms
    const int E  = in_sizes[2] / 2;      // 160000 edges
    const int nf = N * FEAT;

    // ---- workspace carve-up (floats) ----
    float* ws = (float*)d_ws;
    float* s      = ws;  ws += nf;
    float* v      = ws;  ws += 3 * nf;
    float* ds     = ws;  ws += nf;
    float* dv     = ws;  ws += 3 * nf;
    float* phi    = ws;  ws += N * 3 * FEAT;
    float* h      = ws;  ws += nf;           // reused: msg hidden & update hidden
    float* uv     = ws;  ws += 3 * nf;
    float* vv     = ws;  ws += 3 * nf;
    float* sstack = ws;  ws += N * 2 * FEAT;
    float* sp     = ws;  ws += N * 3 * FEAT;
    float* dist   = ws;  ws += E;
    float* unit   = ws;  ws += 3 * E;
    float* env    = ws;  ws += E;

    float* out = (float*)d_out;

    auto blocks = [](int n, int b) { return (n + b - 1) / b; };

    // ---- init ----
    zero_kernel<<<blocks(3 * nf, 256), 256, 0, stream>>>(v, 3 * nf);
    zero_kernel<<<blocks(nf, 256), 256, 0, stream>>>(ds, nf);
    zero_kernel<<<blocks(3 * nf, 256), 256, 0, stream>>>(dv, 3 * nf);
    zero_kernel<<<blocks(out_size, 256), 256, 0, stream>>>(out, out_size);
    embed_kernel<<<blocks(nf, 256), 256, 0, stream>>>(s, embedw, z, N);
    geom_kernel<<<blocks(E, 256), 256, 0, stream>>>(xyz, nbr, dist, unit, env, E);

    // MT=5 register blocking when M divides 80 (10000 = 125*80); else 1 tile/wave.
    auto gemm = [&](const float* A, const float* W, const float* b, float* C,
                    int M, int K, int Nn, int silu, int batch, int as, int cs) {
        if (M % 80 == 0) {
            dim3 g(M / 80, Nn / 16, batch);
            gemm_wmma<5><<<g, 32, 0, stream>>>(A, W, b, C, K, Nn, silu, as, cs);
        } else {
            dim3 g(M / 16, Nn / 16, batch);
            gemm_wmma<1><<<g, 32, 0, stream>>>(A, W, b, C, K, Nn, silu, as, cs);
        }
    };

    for (int l = 0; l < NCONV; ++l) {
        // message MLP (atom-side)
        gemm(s, msg_w1 + l * FEAT * FEAT,     msg_b1 + l * FEAT,     h,   N, FEAT, FEAT,     1, 1, 0, 0);
        gemm(h, msg_w2 + l * FEAT * 3 * FEAT, msg_b2 + l * 3 * FEAT, phi, N, FEAT, 3 * FEAT, 0, 1, 0, 0);

        // per-edge message + atomic segment-sum into ds/dv (L2-resident)
        edge_msg_kernel<<<blocks(E * 32, 256), 256, 0, stream>>>(
            nbr, dist, unit, env, phi, v,
            dist_w + l * NRBF * 3 * FEAT, dist_b + l * 3 * FEAT, ds, dv, E, N);

        apply_kernel<<<blocks(3 * nf, 256), 256, 0, stream>>>(s, v, ds, dv, nf);

        // update block
        gemm(v, upd_u + l * FEAT * FEAT, nullptr, uv, N, FEAT, FEAT, 0, 3, nf, nf);
        gemm(v, upd_v + l * FEAT * FEAT, nullptr, vv, N, FEAT, FEAT, 0, 3, nf, nf);
        stack_kernel<<<blocks(nf, 256), 256, 0, stream>>>(s, vv, sstack, N);
        gemm(sstack, upd_w1 + l * 2 * FEAT * FEAT, upd_b1 + l * FEAT,     h,  N, 2 * FEAT, FEAT,     1, 1, 0, 0);
        gemm(h,      upd_w2 + l * FEAT * 3 * FEAT, upd_b2 + l * 3 * FEAT, sp, N, FEAT,     3 * FEAT, 0, 1, 0, 0);
        updfinal_kernel<<<blocks(nf, 256), 256, 0, stream>>>(s, v, uv, vv, sp, N);
    }

    // readout + force scatter
    edge_force_kernel<<<blocks(E * 32, 128), 128, 0, stream>>>(
        nbr, xyz, s, ero_w1, ero_b1, ero_w2, ero_b2, out, E);
}